// GAttention_66030827208834
// MI455X (gfx1250) — compile-verified
//
#include <hip/hip_runtime.h>

typedef __attribute__((ext_vector_type(2))) float v2f;
typedef __attribute__((ext_vector_type(8))) float v8f;

#define N_NODES 200000
#define N_EDGES 6400000
#define NB      64
#define ROWS    128
#define INF_    128
#define HID     6
#define PADW    8          // padded row stride for xl/xr/num (32 B -> b128 loads)
#define JKO     4
#define FEAT    2560
#define OH      20
#define OUTD    2

// ---------------- float atomic max (ordered-int trick, L2-native atomics) ---
__device__ __forceinline__ void atomicMaxF(float* addr, float v) {
    if (v >= 0.0f)
        atomicMax((int*)addr, __float_as_int(v));
    else
        atomicMin((unsigned int*)addr, (unsigned int)__float_as_int(v));
}

// ---------------- init workspace -------------------------------------------
__global__ void k_init(float* m, float* z, float* num, float* gsum, float* gcnt) {
    int i = blockIdx.x * blockDim.x + threadIdx.x;
    if (i < N_NODES) { m[i] = -__int_as_float(0x7f800000); z[i] = 0.0f; }
    if (i < N_NODES * PADW) num[i] = 0.0f;
    if (i < NB * JKO) gsum[i] = 0.0f;
    if (i < NB) gcnt[i] = 0.0f;
}

// ---------------- fused node transform: xl = x@Wl+bl, xr = x@Wr+br ----------
// One wave per 16-node tile. Combined B: cols 0-5 = Wl, 8-13 = Wr, rest 0,
// so columns 6,7,14,15 produce exact zeros -> unconditional padded stores.
__global__ void k_node_xform(const float* __restrict__ x,
                             const float* __restrict__ Wl, const float* __restrict__ bl,
                             const float* __restrict__ Wr, const float* __restrict__ br,
                             float* __restrict__ xl, float* __restrict__ xr) {
    __shared__ float Bs[INF_ * 16];
    __shared__ float bias_s[16];
    for (int t = threadIdx.x; t < INF_ * 16; t += blockDim.x) {
        int k = t >> 4, n = t & 15;
        float v = 0.0f;
        if (n < HID)                        v = Wl[k * HID + n];
        else if (n >= PADW && n < PADW + HID) v = Wr[k * HID + (n - PADW)];
        Bs[t] = v;
    }
    if (threadIdx.x < 16) {
        int n = threadIdx.x;
        bias_s[n] = (n < HID) ? bl[n]
                  : ((n >= PADW && n < PADW + HID) ? br[n - PADW] : 0.0f);
    }
    __syncthreads();

    int wv   = threadIdx.x >> 5;
    int mt   = blockIdx.x * (blockDim.x >> 5) + wv;       // wave-uniform
    if (mt >= (N_NODES / 16)) return;
    int lane = threadIdx.x & 31;
    int half = lane >> 4, idx = lane & 15;

    v8f c;
    float bs = bias_s[idx];
    #pragma unroll
    for (int r = 0; r < 8; r++) c[r] = bs;

    const float* arow = x + (size_t)(mt * 16 + idx) * INF_ + 2 * half;
    #pragma unroll 4
    for (int k = 0; k < INF_; k += 4) {
        v2f a, b;
        a.x = arow[k];
        a.y = arow[k + 1];
        int kb = k + 2 * half;
        b.x = Bs[kb * 16 + idx];
        b.y = Bs[(kb + 1) * 16 + idx];
        c = __builtin_amdgcn_wmma_f32_16x16x4_f32(false, a, false, b, (short)0, c, false, false);
    }
    int row0 = mt * 16 + 8 * half;
    // cols 0-7 -> xl row (6,7 are exact-zero padding); cols 8-15 -> xr row
    float* base = (idx < PADW) ? xl : xr;
    int col = idx & (PADW - 1);
    #pragma unroll
    for (int r = 0; r < 8; r++)
        base[(size_t)(row0 + r) * PADW + col] = c[r];
}

// ---------------- edge pass 1: scores + segment max -------------------------
__global__ void k_edge_pass1(const int* __restrict__ ei,
                             const float4* __restrict__ xlv, const float4* __restrict__ xrv,
                             const float* __restrict__ att,
                             float* __restrict__ score, float* __restrict__ m) {
    int i = blockIdx.x * blockDim.x + threadIdx.x;
    if (i >= N_EDGES + N_NODES) return;
    int s, d;
    if (i < N_EDGES) { s = ei[i]; d = ei[N_EDGES + i]; }
    else             { s = d = i - N_EDGES; }            // self loop
    float4 a0 = xlv[(size_t)s * 2], a1 = xlv[(size_t)s * 2 + 1];
    float4 b0 = xrv[(size_t)d * 2], b1 = xrv[(size_t)d * 2 + 1];
    float e[HID] = { a0.x + b0.x, a0.y + b0.y, a0.z + b0.z,
                     a0.w + b0.w, a1.x + b1.x, a1.y + b1.y };
    float sc = 0.0f;
    #pragma unroll
    for (int c = 0; c < HID; c++) {
        float v = (e[c] >= 0.0f) ? e[c] : 0.2f * e[c];    // leaky(0.2)
        sc += v * att[c];
    }
    score[i] = sc;
    atomicMaxF(&m[d], sc);
}

// ---------------- edge pass 2: exp + weighted scatter -----------------------
__global__ void k_edge_pass2(const int* __restrict__ ei,
                             const float4* __restrict__ xlv,
                             const float* __restrict__ score, const float* __restrict__ m,
                             float* __restrict__ z, float* __restrict__ num) {
    int i = blockIdx.x * blockDim.x + threadIdx.x;
    if (i >= N_EDGES + N_NODES) return;
    int s, d;
    if (i < N_EDGES) { s = ei[i]; d = ei[N_EDGES + i]; }
    else             { s = d = i - N_EDGES; }
    float p = __expf(score[i] - m[d]);
    atomicAdd(&z[d], p);
    float4 a0 = xlv[(size_t)s * 2], a1 = xlv[(size_t)s * 2 + 1];
    float xs[HID] = { a0.x, a0.y, a0.z, a0.w, a1.x, a1.y };
    float* nrow = num + (size_t)d * PADW;
    #pragma unroll
    for (int c = 0; c < HID; c++)
        atomicAdd(&nrow[c], p * xs[c]);
}

// ---------------- node finalize: agg/z + leaky + JK proj + graph pool -------
__global__ void k_node_fin(const float* __restrict__ numv, const float* __restrict__ z,
                           const float* __restrict__ conv_b,
                           const float* __restrict__ jkW, const float* __restrict__ jkb,
                           const int* __restrict__ batch,
                           float* __restrict__ gsum, float* __restrict__ gcnt) {
    int i = blockIdx.x * blockDim.x + threadIdx.x;
    if (i >= N_NODES) return;
    float inv = 1.0f / z[i];                              // z >= exp(0) from self-loop
    float x1[HID];
    #pragma unroll
    for (int c = 0; c < HID; c++) {
        float a = numv[(size_t)i * PADW + c] * inv + conv_b[c];
        x1[c] = (a >= 0.0f) ? a : 0.01f * a;              // leaky(0.01)
    }
    int b = batch[i];
    #pragma unroll
    for (int j = 0; j < JKO; j++) {
        float h = jkb[j];
        #pragma unroll
        for (int c = 0; c < HID; c++) h += x1[c] * jkW[c * JKO + j];
        atomicAdd(&gsum[b * JKO + j], h);
    }
    atomicAdd(&gcnt[b], 1.0f);
}

__global__ void k_pool_fin(const float* __restrict__ gsum, const float* __restrict__ gcnt,
                           float* __restrict__ xgcn) {
    int i = threadIdx.x;
    if (i < NB * JKO) {
        float c = gcnt[i >> 2];
        xgcn[i] = gsum[i] / fmaxf(c, 1.0f);
    }
}

// ---------------- generic fp32 WMMA GEMM: C = A[MxK] @ B[KxN] + bias --------
__global__ void k_wmma_gemm(const float* __restrict__ A, const float* __restrict__ B,
                            const float* __restrict__ bias, float* __restrict__ C,
                            int M, int K, int Ncols) {
    int lane = threadIdx.x & 31;
    int wv   = threadIdx.x >> 5;
    int ntiles = (Ncols + 15) >> 4;
    int mtiles = M >> 4;
    int wid  = blockIdx.x * (blockDim.x >> 5) + wv;       // wave-uniform
    if (wid >= mtiles * ntiles) return;
    int mt = wid / ntiles, nt = wid - mt * ntiles;
    int half = lane >> 4, idx = lane & 15;
    int n = nt * 16 + idx;
    bool nok = n < Ncols;

    v8f c;
    float bs = nok ? bias[n] : 0.0f;
    #pragma unroll
    for (int r = 0; r < 8; r++) c[r] = bs;

    const float* arow = A + (size_t)(mt * 16 + idx) * K + 2 * half;
    for (int k = 0; k < K; k += 4) {
        v2f a, b;
        a.x = arow[k];
        a.y = arow[k + 1];
        int kb = k + 2 * half;
        b.x = nok ? B[(size_t)kb * Ncols + n] : 0.0f;
        b.y = nok ? B[(size_t)(kb + 1) * Ncols + n] : 0.0f;
        c = __builtin_amdgcn_wmma_f32_16x16x4_f32(false, a, false, b, (short)0, c, false, false);
    }
    if (nok) {
        int row0 = mt * 16 + 8 * half;
        #pragma unroll
        for (int r = 0; r < 8; r++)
            C[(size_t)(row0 + r) * Ncols + n] = c[r];
    }
}

// ---------------- in-place LayerNorm + leaky, one wave per row --------------
__global__ void k_ln_leaky(float* __restrict__ data,
                           const float* __restrict__ g, const float* __restrict__ b,
                           int rows, int D, float slope) {
    int lane = threadIdx.x & 31;
    int row  = blockIdx.x * (blockDim.x >> 5) + (threadIdx.x >> 5);
    if (row >= rows) return;
    float* p = data + (size_t)row * D;
    float sum = 0.0f, sq = 0.0f;
    for (int j = lane; j < D; j += 32) {
        float v = p[j];
        sum += v; sq += v * v;
    }
    #pragma unroll
    for (int off = 16; off > 0; off >>= 1) {
        sum += __shfl_xor(sum, off);
        sq  += __shfl_xor(sq, off);
    }
    float mu = sum / (float)D;
    float var = sq / (float)D - mu * mu;
    float rs = rsqrtf(var + 1e-5f);
    for (int j = lane; j < D; j += 32) {
        float v = (p[j] - mu) * rs * g[j] + b[j];
        p[j] = (v >= 0.0f) ? v : slope * v;
    }
}

// ---------------- final: cat [xgcn|xf|one_hot] @ out_W + out_b --------------
__global__ void k_final(const float* __restrict__ xgcn, const float* __restrict__ xf,
                        const float* __restrict__ oh,
                        const float* __restrict__ oW, const float* __restrict__ ob,
                        float* __restrict__ out) {
    int t = threadIdx.x;
    if (t >= NB * OUTD) return;
    int b = t / OUTD, o = t - b * OUTD;
    float acc = ob[o];
    #pragma unroll
    for (int k = 0; k < JKO; k++) acc += xgcn[b * JKO + k] * oW[k * OUTD + o];
    #pragma unroll 8
    for (int k = 0; k < 32; k++)  acc += xf[b * 32 + k] * oW[(JKO + k) * OUTD + o];
    #pragma unroll 4
    for (int k = 0; k < OH; k++)  acc += oh[b * OH + k] * oW[(JKO + 32 + k) * OUTD + o];
    out[t] = acc;
}

extern "C" void kernel_launch(void* const* d_in, const int* in_sizes, int n_in,
                              void* d_out, int out_size, void* d_ws, size_t ws_size,
                              hipStream_t stream) {
    const float* x        = (const float*)d_in[0];
    const int*   ei       = (const int*)  d_in[1];
    const int*   batch    = (const int*)  d_in[2];
    const float* features = (const float*)d_in[3];
    const float* one_hot  = (const float*)d_in[4];
    const float* W_l      = (const float*)d_in[5];
    const float* b_l      = (const float*)d_in[6];
    const float* W_r      = (const float*)d_in[7];
    const float* b_r      = (const float*)d_in[8];
    const float* att      = (const float*)d_in[9];
    const float* conv_b   = (const float*)d_in[10];
    const float* jk_W     = (const float*)d_in[11];
    const float* jk_b     = (const float*)d_in[12];
    const float* l1_W     = (const float*)d_in[13];
    const float* l1_b     = (const float*)d_in[14];
    const float* ln1_g    = (const float*)d_in[15];
    const float* ln1_b    = (const float*)d_in[16];
    const float* l2_W     = (const float*)d_in[17];
    const float* l2_b     = (const float*)d_in[18];
    const float* ln2_g    = (const float*)d_in[19];
    const float* ln2_b    = (const float*)d_in[20];
    const float* l3_W     = (const float*)d_in[21];
    const float* l3_b     = (const float*)d_in[22];
    const float* ln3_g    = (const float*)d_in[23];
    const float* ln3_b    = (const float*)d_in[24];
    const float* fl_W     = (const float*)d_in[25];
    const float* fl_b     = (const float*)d_in[26];
    const float* ln4_g    = (const float*)d_in[27];
    const float* ln4_b    = (const float*)d_in[28];
    const float* out_W    = (const float*)d_in[29];
    const float* out_b    = (const float*)d_in[30];
    float* out = (float*)d_out;

    // ---- workspace carve-up (floats) ----
    float* ws = (float*)d_ws;
    size_t o = 0;
    float* xl    = ws + o; o += (size_t)N_NODES * PADW;    // padded, 16B-aligned rows
    float* xr    = ws + o; o += (size_t)N_NODES * PADW;
    float* score = ws + o; o += (size_t)(N_EDGES + N_NODES);
    float* mval  = ws + o; o += N_NODES;
    float* zval  = ws + o; o += N_NODES;
    float* numv  = ws + o; o += (size_t)N_NODES * PADW;
    float* gsum  = ws + o; o += NB * JKO;
    float* gcnt  = ws + o; o += NB;
    float* xgcn  = ws + o; o += NB * JKO;
    float* f1    = ws + o; o += (size_t)NB * ROWS * 256;   // 8192 x 256
    float* f2    = ws + o; o += (size_t)NB * ROWS * 64;    // 8192 x 64
    float* f3    = ws + o; o += (size_t)NB * ROWS * 8;     // 8192 x 8 == 64 x 1024
    float* xf    = ws + o; o += NB * 32;

    const int MR = NB * ROWS;                               // 8192

    // init accumulators
    {
        int n = N_NODES * PADW;
        k_init<<<(n + 255) / 256, 256, 0, stream>>>(mval, zval, numv, gsum, gcnt);
    }
    // fused node transform via fp32 WMMA
    {
        int mtiles = N_NODES / 16;                          // 12500
        int blocks = (mtiles + 7) / 8;
        k_node_xform<<<blocks, 256, 0, stream>>>(x, W_l, b_l, W_r, b_r, xl, xr);
    }
    // edge phase (incl. self loops), b128 gathers on padded rows
    {
        int n = N_EDGES + N_NODES;
        int blocks = (n + 255) / 256;
        k_edge_pass1<<<blocks, 256, 0, stream>>>(ei, (const float4*)xl, (const float4*)xr,
                                                 att, score, mval);
        k_edge_pass2<<<blocks, 256, 0, stream>>>(ei, (const float4*)xl, score, mval,
                                                 zval, numv);
    }
    k_node_fin<<<(N_NODES + 255) / 256, 256, 0, stream>>>(numv, zval, conv_b, jk_W, jk_b,
                                                          batch, gsum, gcnt);
    k_pool_fin<<<1, 256, 0, stream>>>(gsum, gcnt, xgcn);

    // MLP chain (WMMA fp32 GEMMs + in-place LN/leaky)
    k_wmma_gemm<<<(MR / 16) * 16 / 8, 256, 0, stream>>>(features, l1_W, l1_b, f1, MR, FEAT, 256);
    k_ln_leaky<<<MR / 8, 256, 0, stream>>>(f1, ln1_g, ln1_b, MR, 256, 0.01f);

    k_wmma_gemm<<<(MR / 16) * 4 / 8, 256, 0, stream>>>(f1, l2_W, l2_b, f2, MR, 256, 64);
    k_ln_leaky<<<MR / 8, 256, 0, stream>>>(f2, ln2_g, ln2_b, MR, 64, 0.01f);

    k_wmma_gemm<<<(MR / 16) * 1 / 8, 256, 0, stream>>>(f2, l3_W, l3_b, f3, MR, 64, 8);
    k_ln_leaky<<<MR / 8, 256, 0, stream>>>(f3, ln3_g, ln3_b, MR, 8, 0.01f);

    // flatten [64,128,8] -> [64,1024] is contiguous; fl GEMM: 64x1024 @ 1024x32
    k_wmma_gemm<<<1, 256, 0, stream>>>(f3, fl_W, fl_b, xf, NB, ROWS * 8, 32);
    k_ln_leaky<<<(NB + 7) / 8, 256, 0, stream>>>(xf, ln4_g, ln4_b, NB, 32, 0.01f);

    k_final<<<1, 128, 0, stream>>>(xgcn, xf, one_hot, out_W, out_b, out);
}